// GraphConv_59322088292606
// MI455X (gfx1250) — compile-verified
//
#include <hip/hip_runtime.h>

typedef __attribute__((ext_vector_type(2))) float v2f;
typedef __attribute__((ext_vector_type(8))) float v8f;

#define INFEAT 128
#define OUTFEAT 128

// ---------------------------------------------------------------------------
// Phase 1: edge scatter. One wave32 per edge. Lane L handles features
// [4L, 4L+4) of the 128-float source row: one float4 load + 4 f32 atomics
// into the per-destination accumulator (L2-resident, 25.6 MB). Lane 0
// accumulates the in-degree count.
// ---------------------------------------------------------------------------
__global__ __launch_bounds__(256) void graphconv_scatter(
    const float* __restrict__ feat,
    const int*   __restrict__ esrc,
    const int*   __restrict__ edst,
    float*       __restrict__ agg,   // [N,128] running sums
    float*       __restrict__ cnt,   // [N] degree counts
    int n_edges)
{
    const int lane = threadIdx.x & 31;
    const int edge = (int)(blockIdx.x * (blockDim.x >> 5)) + (threadIdx.x >> 5);
    if (edge >= n_edges) return;

    const int s = esrc[edge];
    const int d = edst[edge];

    const float4 v = *(const float4*)(feat + (size_t)s * INFEAT + lane * 4);
    float* base = agg + (size_t)d * INFEAT + lane * 4;
    atomicAdd(base + 0, v.x);
    atomicAdd(base + 1, v.y);
    atomicAdd(base + 2, v.z);
    atomicAdd(base + 3, v.w);
    if (lane == 0) atomicAdd(cnt + d, 1.0f);
}

// ---------------------------------------------------------------------------
// Phase 2: fused dual GEMM + mean + concat + ReLU using V_WMMA_F32_16X16X4_F32.
//
// Block = 512 threads (16 waves), 16 output rows per block.
// Wave w: half = w>>3 (0 -> features@W, 1 -> mean(agg)@W), ntile = w&7.
// Weight (64 KB) staged in LDS; per-row mean reciprocals staged in LDS and
// applied in the epilogue (scaling A rows == scaling D rows), so the inner
// loop is pure load + WMMA. Two accumulators break the WMMA RAW chain.
//
// f32 WMMA operand layout (ISA 7.12.2, 32-bit A 16x4 / B 4x16 / C 16x16):
//   A: lane holds row M=lane%16; VGPR0 = K=2*(lane/16), VGPR1 = K+1
//   B: lane holds col N=lane%16; VGPR0 = row K=2*(lane/16), VGPR1 = K+1
//   C/D: VGPR v: lanes 0-15 -> (M=v, N=lane), lanes 16-31 -> (M=v+8, N=lane-16)
// ---------------------------------------------------------------------------
__global__ __launch_bounds__(512) void graphconv_gemm(
    const float* __restrict__ feat,  // [N,128]
    const float* __restrict__ agg,   // [N,128] sums from phase 1
    const float* __restrict__ cnt,   // [N]
    const float* __restrict__ W,     // [128,128]
    float*       __restrict__ out,   // [N,256]
    int n_nodes)
{
    __shared__ float ldsW[INFEAT * OUTFEAT];   // 64 KB (WGP has 320 KB)
    __shared__ float ldsS[16];                 // per-row 1/max(cnt,1)

    const int row0 = blockIdx.x << 4;          // 16 rows per block

    for (int i = threadIdx.x; i < INFEAT * OUTFEAT; i += blockDim.x)
        ldsW[i] = W[i];
    if (threadIdx.x < 16) {
        const int r = row0 + threadIdx.x;
        const float c = (r < n_nodes) ? cnt[r] : 1.0f;
        ldsS[threadIdx.x] = 1.0f / fmaxf(c, 1.0f);
    }
    __syncthreads();

    const int lane  = threadIdx.x & 31;
    const int wave  = threadIdx.x >> 5;
    const int half  = wave >> 3;        // 0: self, 1: aggregated
    const int ntile = wave & 7;         // 16-col tile within the half

    const int m  = lane & 15;           // A row / B,D column within tile
    const int kk = lane >> 4;           // 0 or 1 -> K offset 0 / 2
    const int row = row0 + m;

    // Clamp for address safety only; out-of-range rows produce garbage that
    // lands exclusively in output rows guarded at the store.
    const int rowc = (row < n_nodes) ? row : (n_nodes - 1);
    const float* __restrict__ A = half ? agg : feat;
    const float* Arow = A + (size_t)rowc * INFEAT;
    const int ncol = (ntile << 4) | m;  // column in W / half-output, 0..127

    v8f acc0 = {};
    v8f acc1 = {};
    #pragma unroll
    for (int k = 0; k < INFEAT; k += 8) {
        const int k0 = k + (kk << 1);
        v2f a0, b0;
        a0.x = Arow[k0];
        a0.y = Arow[k0 + 1];
        b0.x = ldsW[k0 * OUTFEAT + ncol];
        b0.y = ldsW[(k0 + 1) * OUTFEAT + ncol];
        acc0 = __builtin_amdgcn_wmma_f32_16x16x4_f32(
            false, a0, false, b0, (short)0, acc0, false, false);

        const int k1 = k0 + 4;
        v2f a1, b1;
        a1.x = Arow[k1];
        a1.y = Arow[k1 + 1];
        b1.x = ldsW[k1 * OUTFEAT + ncol];
        b1.y = ldsW[(k1 + 1) * OUTFEAT + ncol];
        acc1 = __builtin_amdgcn_wmma_f32_16x16x4_f32(
            false, a1, false, b1, (short)0, acc1, false, false);
    }

    // Epilogue: combine accumulators, apply segment-mean scale (half 1 only),
    // fused ReLU, write the concatenated [N,256] output.
    const int ocol = (half << 7) + ncol;            // 0..255
    float* outp = out + (size_t)row0 * (2 * OUTFEAT) + ocol;
    #pragma unroll
    for (int v = 0; v < 8; ++v) {
        const int r = v + (kk << 3);                // D row within block
        if (row0 + r < n_nodes) {
            float val = acc0[v] + acc1[v];
            if (half) val *= ldsS[r];
            outp[(size_t)r * (2 * OUTFEAT)] = fmaxf(val, 0.0f);
        }
    }
}

// ---------------------------------------------------------------------------
// Launch: zero workspace -> scatter -> fused GEMM. All async on `stream`.
// ---------------------------------------------------------------------------
extern "C" void kernel_launch(void* const* d_in, const int* in_sizes, int n_in,
                              void* d_out, int out_size, void* d_ws, size_t ws_size,
                              hipStream_t stream) {
    const float* feat = (const float*)d_in[0];
    const float* W    = (const float*)d_in[1];
    const int*   esrc = (const int*)d_in[2];
    const int*   edst = (const int*)d_in[3];
    float* out = (float*)d_out;

    const int n_nodes = in_sizes[0] / INFEAT;
    const int n_edges = in_sizes[2];

    float* agg = (float*)d_ws;                          // [N,128]
    float* cnt = agg + (size_t)n_nodes * INFEAT;        // [N]
    const size_t zbytes = ((size_t)n_nodes * INFEAT + (size_t)n_nodes) * sizeof(float);
    hipMemsetAsync(d_ws, 0, zbytes, stream);

    const int waves_per_block = 256 / 32;
    const int sblocks = (n_edges + waves_per_block - 1) / waves_per_block;
    graphconv_scatter<<<sblocks, 256, 0, stream>>>(feat, esrc, edst, agg, cnt, n_edges);

    const int gblocks = (n_nodes + 15) / 16;
    graphconv_gemm<<<gblocks, 512, 0, stream>>>(feat, agg, cnt, W, out, n_nodes);
}